// GATLayer_62431644614833
// MI455X (gfx1250) — compile-verified
//
#include <hip/hip_runtime.h>
#include <math.h>

// GAT-style layer for MI455X (gfx1250, wave32).
// Phases:
//   1) h_src = feat_src @ u            (WMMA f32 16x16x4)
//   2) base  = feat_dst @ u,  h_dst = feat_dst @ v   (dual WMMA, shared A-fragments)
//   3) edge gather -> sigmoid gate -> atomic scatter into base (L2-resident)
//   4) LayerNorm over C=64 + residual  (one wave per row, shfl reductions)

typedef __attribute__((ext_vector_type(2))) float v2f;
typedef __attribute__((ext_vector_type(8))) float v8f;

#define CDIM 64
#define TDIM 4
#define ROWLEN 64            // inner dim of [M,64] matrices
#define TCP 256              // T*C per node

// ---------------------------------------------------------------------------
// WMMA helpers: one wave computes a 16x16 f32 tile with K=64 accumulation.
// A-fragment layout (16x4 f32): lanes 0-15 -> K=0,1 ; lanes 16-31 -> K=2,3.
// B-fragment layout (4x16 f32): same K split, N = lane&15.
// D layout (16x16 f32, 8 VGPRs): VGPR i -> row i (lanes 0-15) / row i+8 (16-31).
// ---------------------------------------------------------------------------
__device__ __forceinline__ v8f wmma16x16_k64(const float* __restrict__ Arows,
                                             const float* __restrict__ Blds,
                                             int c0) {
    const int lane = threadIdx.x & 31;
    const int m    = lane & 15;
    const int koff = (lane >> 4) << 1;          // 0 or 2
    const float* __restrict__ arow = Arows + m * ROWLEN;
    v8f acc = {0.f, 0.f, 0.f, 0.f, 0.f, 0.f, 0.f, 0.f};
#pragma unroll
    for (int k0 = 0; k0 < 64; k0 += 4) {
        v2f a, b;
        a.x = arow[k0 + koff];
        a.y = arow[k0 + koff + 1];
        b.x = Blds[(k0 + koff)     * ROWLEN + c0 + m];
        b.y = Blds[(k0 + koff + 1) * ROWLEN + c0 + m];
        acc = __builtin_amdgcn_wmma_f32_16x16x4_f32(
            false, a, false, b, (short)0, acc, false, false);
    }
    return acc;
}

__device__ __forceinline__ void store_tile(float* __restrict__ Out, int row0,
                                           int c0, v8f acc) {
    const int lane = threadIdx.x & 31;
    const int col  = c0 + (lane & 15);
    const int rb   = row0 + ((lane >> 4) << 3);
#pragma unroll
    for (int i = 0; i < 8; ++i)
        Out[(rb + i) * ROWLEN + col] = acc[i];
}

// ----------------------------- phase 1: h_src = feat_src @ u ---------------
__global__ void __launch_bounds__(128)
proj_src_kernel(const float* __restrict__ feat, const float* __restrict__ U,
                float* __restrict__ Out, int mtiles) {
    __shared__ float ldsU[CDIM * CDIM];
    for (int i = threadIdx.x; i < CDIM * CDIM; i += blockDim.x) ldsU[i] = U[i];
    __syncthreads();

    const int wave = threadIdx.x >> 5;
    const int tile = blockIdx.x * (blockDim.x >> 5) + wave;
    if (tile >= mtiles) return;                 // wave-uniform: EXEC stays all-1s
    const int row0 = tile * 16;
    const float* __restrict__ Arows = feat + (long long)row0 * ROWLEN;
#pragma unroll
    for (int c0 = 0; c0 < 64; c0 += 16) {
        v8f acc = wmma16x16_k64(Arows, ldsU, c0);
        store_tile(Out, row0, c0, acc);
    }
}

// ------------- phase 2: base = feat_dst @ u ; h_dst = feat_dst @ v ---------
__global__ void __launch_bounds__(128)
proj_dst_kernel(const float* __restrict__ feat, const float* __restrict__ U,
                const float* __restrict__ V, float* __restrict__ OutBase,
                float* __restrict__ OutV, int mtiles) {
    __shared__ float ldsU[CDIM * CDIM];
    __shared__ float ldsV[CDIM * CDIM];
    for (int i = threadIdx.x; i < CDIM * CDIM; i += blockDim.x) {
        ldsU[i] = U[i];
        ldsV[i] = V[i];
    }
    __syncthreads();

    const int wave = threadIdx.x >> 5;
    const int tile = blockIdx.x * (blockDim.x >> 5) + wave;
    if (tile >= mtiles) return;
    const int row0 = tile * 16;
    const int lane = threadIdx.x & 31;
    const int m    = lane & 15;
    const int koff = (lane >> 4) << 1;
    const float* __restrict__ arow = feat + ((long long)row0 + m) * ROWLEN;

#pragma unroll
    for (int c0 = 0; c0 < 64; c0 += 16) {
        v8f accU = {0.f, 0.f, 0.f, 0.f, 0.f, 0.f, 0.f, 0.f};
        v8f accV = {0.f, 0.f, 0.f, 0.f, 0.f, 0.f, 0.f, 0.f};
#pragma unroll
        for (int k0 = 0; k0 < 64; k0 += 4) {
            v2f a, bu, bv;
            a.x  = arow[k0 + koff];
            a.y  = arow[k0 + koff + 1];
            bu.x = ldsU[(k0 + koff)     * ROWLEN + c0 + m];
            bu.y = ldsU[(k0 + koff + 1) * ROWLEN + c0 + m];
            bv.x = ldsV[(k0 + koff)     * ROWLEN + c0 + m];
            bv.y = ldsV[(k0 + koff + 1) * ROWLEN + c0 + m];
            accU = __builtin_amdgcn_wmma_f32_16x16x4_f32(
                false, a, false, bu, (short)0, accU, false, false);
            accV = __builtin_amdgcn_wmma_f32_16x16x4_f32(
                false, a, false, bv, (short)0, accV, false, false);
        }
        store_tile(OutBase, row0, c0, accU);
        store_tile(OutV,    row0, c0, accV);
    }
}

// --------------------- phase 3: per-edge gate + atomic scatter -------------
// One thread = 4 contiguous channels of one (edge, t) row. Gathers are float4
// (16B/lane, coalesced within an edge); tables are L2-resident (10 MB each).
__global__ void __launch_bounds__(256)
edge_kernel(const float* __restrict__ h_src, const float* __restrict__ h_dst,
            const float* __restrict__ edge_w, const float* __restrict__ we,
            const int* __restrict__ src_idx, const int* __restrict__ dst_idx,
            float* __restrict__ aggr, long long total) {
    long long i = (long long)blockIdx.x * blockDim.x + threadIdx.x;
    if (i >= total) return;
    const int e   = (int)(i >> 6);        // 64 quads per edge (T*C/4)
    const int r   = (int)(i & 63);
    const int tc4 = r << 2;               // offset within [T*C]=256 floats
    const int c   = tc4 & (CDIM - 1);     // channel within C

    const int   s = src_idx[e];
    const int   d = dst_idx[e];
    const float w = edge_w[e];

    const float4 hs = *(const float4*)(h_src + (long long)s * TCP + tc4);
    const float4 hd = *(const float4*)(h_dst + (long long)d * TCP + tc4);
    const float4 wv = *(const float4*)(we + c);
    float* __restrict__ out = aggr + (long long)d * TCP + tc4;

    {
        float x = hs.x * hd.x * (w * wv.x);
        atomicAdd(out + 0, hs.x / (1.0f + __expf(-x)));
    }
    {
        float x = hs.y * hd.y * (w * wv.y);
        atomicAdd(out + 1, hs.y / (1.0f + __expf(-x)));
    }
    {
        float x = hs.z * hd.z * (w * wv.z);
        atomicAdd(out + 2, hs.z / (1.0f + __expf(-x)));
    }
    {
        float x = hs.w * hd.w * (w * wv.w);
        atomicAdd(out + 3, hs.w / (1.0f + __expf(-x)));
    }
}

// -------------------- phase 4: LayerNorm(C=64) + residual ------------------
// One wave32 per row; each lane owns channels (lane, lane+32).
__global__ void __launch_bounds__(256)
ln_kernel(const float* __restrict__ acc, const float* __restrict__ feat_dst,
          const float* __restrict__ gamma, const float* __restrict__ beta,
          float* __restrict__ out, int rows) {
    const int wave = threadIdx.x >> 5;
    const int lane = threadIdx.x & 31;
    const int row  = blockIdx.x * (blockDim.x >> 5) + wave;
    if (row >= rows) return;

    const float* __restrict__ x = acc + (long long)row * CDIM;
    float x0 = x[lane];
    float x1 = x[lane + 32];

    float s = x0 + x1;
#pragma unroll
    for (int off = 16; off > 0; off >>= 1) s += __shfl_xor(s, off, 32);
    const float mean = s * (1.0f / 64.0f);

    const float d0 = x0 - mean, d1 = x1 - mean;
    float q = d0 * d0 + d1 * d1;
#pragma unroll
    for (int off = 16; off > 0; off >>= 1) q += __shfl_xor(q, off, 32);
    const float rstd = rsqrtf(q * (1.0f / 64.0f) + 1e-5f);

    const float* __restrict__ f = feat_dst + (long long)row * CDIM;
    out[(long long)row * CDIM + lane] =
        d0 * rstd * gamma[lane] + beta[lane] + f[lane];
    out[(long long)row * CDIM + lane + 32] =
        d1 * rstd * gamma[lane + 32] + beta[lane + 32] + f[lane + 32];
}

// ---------------------------------------------------------------------------
extern "C" void kernel_launch(void* const* d_in, const int* in_sizes, int n_in,
                              void* d_out, int out_size, void* d_ws, size_t ws_size,
                              hipStream_t stream) {
    const float* feat_src = (const float*)d_in[0];   // [N_SRC, T, C]
    const float* feat_dst = (const float*)d_in[1];   // [N_DST, T, C]
    const float* edge_w   = (const float*)d_in[2];   // [E]
    const float* weight_e = (const float*)d_in[3];   // [1, C]
    const float* u        = (const float*)d_in[4];   // [C, C]
    const float* v        = (const float*)d_in[5];   // [C, C]
    const float* ln_gamma = (const float*)d_in[6];   // [C]
    const float* ln_beta  = (const float*)d_in[7];   // [C]
    const int*   src_idx  = (const int*)d_in[8];     // [E]
    const int*   dst_idx  = (const int*)d_in[9];     // [E]
    float*       out      = (float*)d_out;

    const int n_src = in_sizes[0] / TCP;             // 10000
    const int n_dst = in_sizes[1] / TCP;             // 10000
    const int E     = in_sizes[2];                   // 320000

    // Workspace: h_src | h_dst | aggr (each [N, T, C] f32, ~10.24 MB -> L2-resident)
    float* ws    = (float*)d_ws;
    float* h_src = ws;
    float* h_dst = ws + (size_t)n_src * TCP;
    float* aggr  = h_dst + (size_t)n_dst * TCP;

    const int mtiles_src = (n_src * TDIM) / 16;      // 2500
    const int mtiles_dst = (n_dst * TDIM) / 16;      // 2500
    const int wpb = 4;                               // 128 threads = 4 wave32

    proj_src_kernel<<<(mtiles_src + wpb - 1) / wpb, 128, 0, stream>>>(
        feat_src, u, h_src, mtiles_src);

    proj_dst_kernel<<<(mtiles_dst + wpb - 1) / wpb, 128, 0, stream>>>(
        feat_dst, u, v, aggr, h_dst, mtiles_dst);

    const long long edge_work = (long long)E * (TCP / 4);   // 20.48M quads
    edge_kernel<<<(int)((edge_work + 255) / 256), 256, 0, stream>>>(
        h_src, h_dst, edge_w, weight_e, src_idx, dst_idx, aggr, edge_work);

    const int rows = n_dst * TDIM;                   // 40000
    ln_kernel<<<(rows + 7) / 8, 256, 0, stream>>>(
        aggr, feat_dst, ln_gamma, ln_beta, out, rows);
}